// DisentanglementModule_47794396070519
// MI455X (gfx1250) — compile-verified
//
#include <hip/hip_runtime.h>
#include <hip/hip_bf16.h>

#define TAU_POS 0.1f
#define TAU_NEG 0.1f

typedef __attribute__((ext_vector_type(2))) float v2f;
typedef __attribute__((ext_vector_type(8))) float v8f;

// ---------------------------------------------------------------------------
// Kernel 1: class histogram -> c_pos / c_neg, and zero the pos/neg accumulators
// ws[0]=c_pos, ws[1]=c_neg, ws[2]=pos_acc, ws[3]=neg_acc
// ---------------------------------------------------------------------------
__global__ void count_kernel(const int* __restrict__ levels, int B, float* __restrict__ ws) {
    __shared__ int hist[4];
    if (threadIdx.x < 4) hist[threadIdx.x] = 0;
    __syncthreads();
    for (int i = threadIdx.x; i < B; i += blockDim.x) {
        atomicAdd(&hist[levels[i] & 3], 1);
    }
    __syncthreads();
    if (threadIdx.x == 0) {
        long long cpos = 0;
        for (int l = 0; l < 4; ++l) {
            long long n = hist[l];
            cpos += n * (n - 1) / 2;
        }
        long long total = (long long)B * (B - 1) / 2;
        ws[0] = (float)cpos;
        ws[1] = (float)(total - cpos);
        ws[2] = 0.0f;
        ws[3] = 0.0f;
    }
}

// ---------------------------------------------------------------------------
// Kernel 2: per-row squared norms (one block per row)
// ---------------------------------------------------------------------------
__global__ void rowsq_kernel(const float* __restrict__ X, int ld, int dim,
                             float* __restrict__ out) {
    const int row = blockIdx.x;
    const float* p = X + (size_t)row * ld;
    float s = 0.0f;
    for (int k = threadIdx.x; k < dim; k += blockDim.x) {
        float x = p[k];
        s += x * x;
    }
    __shared__ float sm[256];
    sm[threadIdx.x] = s;
    __syncthreads();
    for (int st = 128; st > 0; st >>= 1) {
        if (threadIdx.x < st) sm[threadIdx.x] += sm[threadIdx.x + st];
        __syncthreads();
    }
    if (threadIdx.x == 0) out[row] = sm[0];
}

// ---------------------------------------------------------------------------
// Kernel 3: fused X·X^T (fp32 WMMA) + triangular contrastive reduction.
// Grid: (B/16 j-tiles, B/128 i-blocks); 256 threads = 8 waves; 16x16 tile/wave.
// V_WMMA_F32_16X16X4_F32 layouts:
//   A (16x4):  lane l<16 holds row M=l, K=(0,1); lane l+16 holds row M=l, K=(2,3)
//   B (4x16):  symmetric -> for G = X X^T both operands load as contiguous float2
//   C (16x16): VGPR v, lanes 0-15: (M=v, N=lane); lanes 16-31: (M=v+8, N=lane-16)
// ---------------------------------------------------------------------------
__global__ void pair_wmma_kernel(const float* __restrict__ X, int ld, int dim,
                                 const float* __restrict__ sq,
                                 const int* __restrict__ levels,
                                 float weight, float* __restrict__ acc) {
    const int jbase  = blockIdx.x * 16;
    const int iblock = blockIdx.y * 128;
    // whole block strictly below the diagonal: no i<j pair exists
    if (iblock >= jbase + 15) return;

    const int tid  = threadIdx.x;
    const int wid  = tid >> 5;       // wave id 0..7
    const int lane = tid & 31;
    const int half = lane >> 4;      // 0 or 1
    const int l15  = lane & 15;

    const int ibase = iblock + wid * 16;

    float pos = 0.0f, neg = 0.0f;

    if (ibase < jbase + 15) {        // wave-uniform: EXEC stays all-ones for WMMA
        const int rowA = ibase + l15;
        const int rowB = jbase + l15;
        const float* pa = X + (size_t)rowA * ld + 2 * half;
        const float* pb = X + (size_t)rowB * ld + 2 * half;

        v8f c0 = {0.f,0.f,0.f,0.f,0.f,0.f,0.f,0.f};
        v8f c1 = {0.f,0.f,0.f,0.f,0.f,0.f,0.f,0.f};

        for (int kb = 0; kb < dim; kb += 64) {
            // warm L2/L0 ahead of the streaming K dimension (global_prefetch_b8)
            __builtin_prefetch(pa + kb + 256, 0, 0);
            __builtin_prefetch(pb + kb + 256, 0, 0);
#pragma unroll
            for (int ku = 0; ku < 64; ku += 8) {
                const int k = kb + ku;
                v2f a0 = *(const v2f*)(pa + k);
                v2f b0 = *(const v2f*)(pb + k);
                c0 = __builtin_amdgcn_wmma_f32_16x16x4_f32(
                        false, a0, false, b0, (short)0, c0, false, false);
                v2f a1 = *(const v2f*)(pa + k + 4);
                v2f b1 = *(const v2f*)(pb + k + 4);
                c1 = __builtin_amdgcn_wmma_f32_16x16x4_f32(
                        false, a1, false, b1, (short)0, c1, false, false);
            }
        }

        // Epilogue: this lane owns column j = jbase + l15, rows i = ibase + v + 8*half
        const int   j       = jbase + l15;
        const float sqj     = sq[j];
        const int   lvlj    = levels[j];
        const float inv_dim = 1.0f / (float)dim;
#pragma unroll
        for (int v = 0; v < 8; ++v) {
            const int i = ibase + v + 8 * half;
            if (i < j) {
                const float g = c0[v] + c1[v];
                const float d = (sq[i] + sqj - 2.0f * g) * inv_dim;
                if (levels[i] == lvlj) pos += d - TAU_POS;
                else                   neg += TAU_NEG - d;
            }
        }
    }

    // block reduction, then one float atomic per accumulator per block
    __shared__ float sp[256];
    __shared__ float sn[256];
    sp[tid] = pos;
    sn[tid] = neg;
    __syncthreads();
    for (int st = 128; st > 0; st >>= 1) {
        if (tid < st) {
            sp[tid] += sp[tid + st];
            sn[tid] += sn[tid + st];
        }
        __syncthreads();
    }
    if (tid == 0) {
        atomicAdd(&acc[0], sp[0] * weight);
        atomicAdd(&acc[1], sn[0] * weight);
    }
}

// ---------------------------------------------------------------------------
// Kernel 4: finalize scalar
// ---------------------------------------------------------------------------
__global__ void finalize_kernel(const float* __restrict__ ws, float* __restrict__ out) {
    out[0] = ws[2] / ws[0] + ws[3] / ws[1];
}

// ---------------------------------------------------------------------------
extern "C" void kernel_launch(void* const* d_in, const int* in_sizes, int n_in,
                              void* d_out, int out_size, void* d_ws, size_t ws_size,
                              hipStream_t stream) {
    (void)in_sizes; (void)n_in; (void)out_size; (void)ws_size;

    const float* rep = (const float*)d_in[0];   // [4096, 1024]
    const float* a0  = (const float*)d_in[1];   // [4096, 512] -> use [:, :64]
    const float* a1  = (const float*)d_in[2];   // [4096, 512] -> use [:, :128]
    const float* a2  = (const float*)d_in[3];   // [4096, 512] -> use [:, :256]
    const int*   lev = (const int*)d_in[4];     // [4096]
    float* out = (float*)d_out;
    float* ws  = (float*)d_ws;

    const int B = 4096;
    float* sq0 = ws + 16;
    float* sq1 = sq0 + B;
    float* sq2 = sq1 + B;
    float* sq3 = sq2 + B;

    count_kernel<<<1, 256, 0, stream>>>(lev, B, ws);

    rowsq_kernel<<<B, 256, 0, stream>>>(rep, 1024, 1024, sq0);
    rowsq_kernel<<<B, 256, 0, stream>>>(a0,  512,  64,  sq1);
    rowsq_kernel<<<B, 256, 0, stream>>>(a1,  512,  128, sq2);
    rowsq_kernel<<<B, 256, 0, stream>>>(a2,  512,  256, sq3);

    dim3 grid(B / 16, B / 128);
    const float w_out = 0.5f;          // (1 - W)
    const float w_lat = 0.5f / 3.0f;   // W / len(LSIZES)

    pair_wmma_kernel<<<grid, 256, 0, stream>>>(rep, 1024, 1024, sq0, lev, w_out, ws + 2);
    pair_wmma_kernel<<<grid, 256, 0, stream>>>(a0,  512,  64,  sq1, lev, w_lat, ws + 2);
    pair_wmma_kernel<<<grid, 256, 0, stream>>>(a1,  512,  128, sq2, lev, w_lat, ws + 2);
    pair_wmma_kernel<<<grid, 256, 0, stream>>>(a2,  512,  256, sq3, lev, w_lat, ws + 2);

    finalize_kernel<<<1, 1, 0, stream>>>(ws, out);
}